// PFNet_56599079026972
// MI455X (gfx1250) — compile-verified
//
#include <hip/hip_runtime.h>
#include <math.h>

// ---- problem constants (from reference) ----
#define NN   50000
#define EE   800000
#define DD   128
#define OUTD 128
#define HH   8
#define DHH  16
#define EPSV 1e-5f

typedef __attribute__((ext_vector_type(2))) float v2f;
typedef __attribute__((ext_vector_type(8))) float v8f;

// ---------------------------------------------------------------
// init: zero out[N*OUT], denom[N*H]; smax[N*H] = -inf
// ---------------------------------------------------------------
__global__ void pf_init(float* __restrict__ out, float* __restrict__ smax,
                        float* __restrict__ denom) {
    int i = blockIdx.x * blockDim.x + threadIdx.x;
    if (i < NN * OUTD) out[i] = 0.0f;
    if (i < NN * HH) {
        smax[i]  = __int_as_float(0xFF800000); // -inf
        denom[i] = 0.0f;
    }
}

// ---------------------------------------------------------------
// Projection GEMM via V_WMMA_F32_16X16X4_F32.
// One wave -> one 16x16 tile. colTile 0..7 = Q head h, 8..15 = V head h.
// Q tiles get per-row LayerNorm across the 16 lanes (DH == tile width).
// ---------------------------------------------------------------
__global__ __launch_bounds__(128)
void pf_proj(const float* __restrict__ x,
             const float* __restrict__ Wq, const float* __restrict__ bq,
             const float* __restrict__ Wv, const float* __restrict__ bv,
             float* __restrict__ Qn, float* __restrict__ Vb) {
    const int lane = threadIdx.x & 31;
    const int wave = threadIdx.x >> 5;
    const int tile = blockIdx.x * 4 + wave;
    const int numTiles = (NN / 16) * 16;           // 3125 row tiles * 16 col tiles
    if (tile >= numTiles) return;                   // uniform per wave

    const int rowTile = tile >> 4;                  // 0..3124
    const int colTile = tile & 15;                  // 0..15
    const bool isQ = (colTile < 8);
    const int h = colTile & 7;
    const int m0 = rowTile * 16;

    const float* W    = (isQ ? Wq : Wv) + (size_t)h * (DD * DHH); // [d][e] row-major
    const float* bias = (isQ ? bq : bv) + h * DHH;

    const int half = lane >> 4;                     // 0: K lo pair, 1: K hi pair
    const int l16  = lane & 15;
    const float* arow = x + (size_t)(m0 + l16) * DD;

    v8f acc = {};
    #pragma unroll 4
    for (int k0 = 0; k0 < DD; k0 += 4) {
        // A[16x4]: lane(l16,half) holds x[m0+l16][k0 + 2*half + {0,1}]
        const float* ap = arow + k0 + 2 * half;
        v2f a; a.x = ap[0]; a.y = ap[1];
        // B[4x16]: lane(l16,half) holds W[k0 + 2*half + {0,1}][l16]
        const float* bp = W + (size_t)(k0 + 2 * half) * DHH + l16;
        v2f b; b.x = bp[0]; b.y = bp[DHH];
        acc = __builtin_amdgcn_wmma_f32_16x16x4_f32(
                  false, a, false, b, (short)0, acc, false, false);
    }

    const float bcol = bias[l16];                   // bias per output column
    #pragma unroll
    for (int r = 0; r < 8; ++r) acc[r] += bcol;

    if (isQ) {
        // LayerNorm each tile row over its 16 columns (one 16-lane group).
        #pragma unroll
        for (int r = 0; r < 8; ++r) {
            float v = acc[r];
            float s = v, s2 = v * v;
            #pragma unroll
            for (int m = 8; m >= 1; m >>= 1) {
                s  += __shfl_xor(s,  m, 16);
                s2 += __shfl_xor(s2, m, 16);
            }
            float mu  = s * (1.0f / 16.0f);
            float var = s2 * (1.0f / 16.0f) - mu * mu;
            acc[r] = (v - mu) * rsqrtf(var + EPSV);
        }
    }

    float* dbuf = isQ ? Qn : Vb;
    #pragma unroll
    for (int r = 0; r < 8; ++r) {
        int row = m0 + r + 8 * half;                // C/D layout: VGPR r, lane half
        dbuf[(size_t)row * OUTD + h * DHH + l16] = acc[r];
    }
}

// ---------------------------------------------------------------
// float atomic-max via signed-max / unsigned-min bit trick
// ---------------------------------------------------------------
__device__ __forceinline__ void atomicMaxF(float* addr, float val) {
    if (val >= 0.0f) atomicMax((int*)addr, __float_as_int(val));
    else             atomicMin((unsigned int*)addr, (unsigned int)__float_as_int(val));
}

// ---------------------------------------------------------------
// edge logits: s[e,h] = dot(Qn[src], Qn[dst]) / OUT ; segment max by src
// ---------------------------------------------------------------
__global__ void pf_edge_logits(const float* __restrict__ Qn,
                               const int* __restrict__ src, const int* __restrict__ dst,
                               float* __restrict__ sbuf, float* __restrict__ smax) {
    int t = blockIdx.x * blockDim.x + threadIdx.x;
    if (t >= EE * HH) return;
    int e = t >> 3, h = t & 7;
    int s = src[e], d = dst[e];
    const float4* qa = (const float4*)(Qn + (size_t)s * OUTD + h * DHH);
    const float4* qb = (const float4*)(Qn + (size_t)d * OUTD + h * DHH);
    float acc = 0.0f;
    #pragma unroll
    for (int i = 0; i < 4; ++i) {
        float4 a = qa[i], b = qb[i];
        acc += a.x * b.x + a.y * b.y + a.z * b.z + a.w * b.w;
    }
    acc *= (1.0f / OUTD);
    sbuf[t] = acc;
    atomicMaxF(&smax[s * HH + h], acc);
}

// ---------------------------------------------------------------
// exp(s - smax[src]) ; segment sum by src
// ---------------------------------------------------------------
__global__ void pf_edge_exp(const int* __restrict__ src,
                            float* __restrict__ sbuf,
                            const float* __restrict__ smax,
                            float* __restrict__ denom) {
    int t = blockIdx.x * blockDim.x + threadIdx.x;
    if (t >= EE * HH) return;
    int e = t >> 3, h = t & 7;
    int s = src[e];
    float v = __expf(sbuf[t] - smax[s * HH + h]);
    sbuf[t] = v;
    atomicAdd(&denom[s * HH + h], v);
}

// ---------------------------------------------------------------
// att = e/denom[src]; scatter-add V[dst]*att into out[src]
// ---------------------------------------------------------------
__global__ void pf_edge_scatter(const int* __restrict__ src, const int* __restrict__ dst,
                                const float* __restrict__ sbuf,
                                const float* __restrict__ denom,
                                const float* __restrict__ Vb,
                                float* __restrict__ out) {
    int t = blockIdx.x * blockDim.x + threadIdx.x;
    if (t >= EE * HH) return;
    int e = t >> 3, h = t & 7;
    int s = src[e], d = dst[e];
    float att = sbuf[t] / denom[s * HH + h];
    const float4* vp = (const float4*)(Vb + (size_t)d * OUTD + h * DHH);
    float* op = out + (size_t)s * OUTD + h * DHH;
    #pragma unroll
    for (int i = 0; i < 4; ++i) {
        float4 v = vp[i];
        atomicAdd(op + 4 * i + 0, v.x * att);
        atomicAdd(op + 4 * i + 1, v.y * att);
        atomicAdd(op + 4 * i + 2, v.z * att);
        atomicAdd(op + 4 * i + 3, v.w * att);
    }
}

// ---------------------------------------------------------------
// BatchNorm1d (eval) + leaky_relu(0.01), in place on out
// ---------------------------------------------------------------
__global__ void pf_bn_act(float* __restrict__ out,
                          const float* __restrict__ gamma, const float* __restrict__ beta,
                          const float* __restrict__ rmean, const float* __restrict__ rvar) {
    int i = blockIdx.x * blockDim.x + threadIdx.x;
    if (i >= NN * OUTD) return;
    int c = i & (OUTD - 1);
    float v = out[i];
    v = (v - rmean[c]) * rsqrtf(rvar[c] + EPSV) * gamma[c] + beta[c];
    out[i] = (v > 0.0f) ? v : 0.01f * v;
}

extern "C" void kernel_launch(void* const* d_in, const int* in_sizes, int n_in,
                              void* d_out, int out_size, void* d_ws, size_t ws_size,
                              hipStream_t stream) {
    const float* x     = (const float*)d_in[0];
    const int*   src   = (const int*)d_in[1];
    const int*   dst   = (const int*)d_in[2];
    const float* Wq    = (const float*)d_in[3];
    const float* bq    = (const float*)d_in[4];
    const float* Wv    = (const float*)d_in[5];
    const float* bv    = (const float*)d_in[6];
    const float* gamma = (const float*)d_in[7];
    const float* beta  = (const float*)d_in[8];
    const float* rmean = (const float*)d_in[9];
    const float* rvar  = (const float*)d_in[10];
    float* out = (float*)d_out;

    // workspace layout (floats): Qn | Vb | sbuf | smax | denom  (~80 MB)
    float* ws    = (float*)d_ws;
    float* Qn    = ws;                         // N*OUT
    float* Vb    = Qn + (size_t)NN * OUTD;     // N*OUT
    float* sbuf  = Vb + (size_t)NN * OUTD;     // E*H
    float* smax  = sbuf + (size_t)EE * HH;     // N*H
    float* denom = smax + (size_t)NN * HH;     // N*H

    const int elemN = NN * OUTD;               // 6.4M
    const int elemE = EE * HH;                 // 6.4M

    pf_init<<<(elemN + 255) / 256, 256, 0, stream>>>(out, smax, denom);

    // 3125 row tiles * 16 col tiles = 50000 waves; 4 waves (128 thr) per block
    const int numTiles = (NN / 16) * 16;
    pf_proj<<<(numTiles + 3) / 4, 128, 0, stream>>>(x, Wq, bq, Wv, bv, Qn, Vb);

    pf_edge_logits<<<(elemE + 255) / 256, 256, 0, stream>>>(Qn, src, dst, sbuf, smax);
    pf_edge_exp<<<(elemE + 255) / 256, 256, 0, stream>>>(src, sbuf, smax, denom);
    pf_edge_scatter<<<(elemE + 255) / 256, 256, 0, stream>>>(src, dst, sbuf, denom, Vb, out);
    pf_bn_act<<<(elemN + 255) / 256, 256, 0, stream>>>(out, gamma, beta, rmean, rvar);
}